// Assessor_26225070309949
// MI455X (gfx1250) — compile-verified
//
#include <hip/hip_runtime.h>

#define HW     (128 * 128)          /* one z-slice                */
#define DHW    (128 * 128 * 128)    /* one batch element (2^21)   */
#define NTOT   (4 * DHW)            /* 8,388,608 voxels           */
#define NSLICE (4 * 128)            /* 512 (b,z) slices           */

typedef __attribute__((ext_vector_type(2))) float        v2f;
typedef __attribute__((ext_vector_type(8))) float        v8f;
typedef __attribute__((ext_vector_type(4))) unsigned int u32x4;
typedef __attribute__((ext_vector_type(8))) int          i32x8;
typedef __attribute__((ext_vector_type(4))) int          i32x4;

// ---------------------------------------------------------------------------
// TDM: DMA a 128x128 f32 tile (one z-slice, 64 KB) from global into LDS.
// D# built per CDNA5 ISA: group0 = count/lds/global/type,
// group1 = data_size + dims/strides. Tracked by TENSORcnt.
// Probe only in the DEVICE pass: the builtin does not exist in the x86 host
// pass, but __device__ bodies are never executed there.
// ---------------------------------------------------------------------------
#if defined(__HIP_DEVICE_COMPILE__)
#if !__has_builtin(__builtin_amdgcn_tensor_load_to_lds)
#error "CDNA5 TDM builtin __builtin_amdgcn_tensor_load_to_lds not available in device pass"
#endif
#define HAVE_TDM 1
#endif

__device__ __forceinline__ void tdm_load_slice(const float* gsrc, unsigned lds_byte) {
#if defined(HAVE_TDM)
    unsigned long long ga = (unsigned long long)gsrc;
    u32x4 g0;
    g0[0] = 1u;                                               // count=1, is_restore=0
    g0[1] = lds_byte;                                         // lds_addr (bytes)
    g0[2] = (unsigned)(ga & 0xFFFFFFFFull);                   // global_addr[31:0]
    g0[3] = (unsigned)((ga >> 32) & 0x01FFFFFFull) | (2u << 30); // addr[56:32], type=2
    i32x8 g1;
    g1[0] = (int)(2u << 16);          // workgroup_mask=0 (not in cluster), data_size=4B
    g1[1] = (int)(128u << 16);        // tensor_dim0 = 128 (low 16 bits)
    g1[2] = (int)(128u << 16);        // tensor_dim1 = 128
    g1[3] = (int)(128u << 16);        // tile_dim0   = 128
    g1[4] = 128;                      // tile_dim1 = 128, tile_dim2 = 0
    g1[5] = 128;                      // tensor_dim0_stride = 128 elements
    g1[6] = (int)(16384u << 16);      // tensor_dim1_stride = 128*128 (low 16 bits)
    g1[7] = 0;
    i32x4 z4 = {0, 0, 0, 0};
#if defined(__clang_major__) && __clang_major__ >= 23
    i32x8 z8 = {0, 0, 0, 0, 0, 0, 0, 0};
    __builtin_amdgcn_tensor_load_to_lds(g0, g1, z4, z4, z8, 0);
#else
    __builtin_amdgcn_tensor_load_to_lds(g0, g1, z4, z4, 0);
#endif
    __builtin_amdgcn_s_wait_tensorcnt(0);
#else
    (void)gsrc; (void)lds_byte;       // host pass stub, never executed
#endif
}

// issued by wave 0 only; all callers barrier afterwards
__device__ __forceinline__ void load_slice_to_lds(float* sm, const float* gsrc, int tid) {
    if (tid < 32) {
        tdm_load_slice(gsrc, (unsigned)(unsigned long long)(void*)sm);
    }
}

// ===========================================================================
// Kernel 1: fused overhang / surface-weight / occupancy partials, 1 block per
// (b,z) slice. surface_term needs no conv: sum(conv3d(x, ones^3, pad=1)) ==
// sum_x x * wz(z)*wy(y)*wx(x) with wd = 2 at boundaries else 3.
// ===========================================================================
__global__ __launch_bounds__(256)
void metrics_kernel(const float* __restrict__ vox,
                    float* __restrict__ p_ovh,
                    float* __restrict__ p_surf,
                    float* __restrict__ p_occ) {
    __shared__ float sm[HW];
    const int s   = blockIdx.x;      // b*128 + z
    const int z   = s & 127;
    const int tid = threadIdx.x;
    const long long base = (long long)s * HW;

    if (z > 0)  // block-uniform
        load_slice_to_lds(sm, vox + base - HW, tid);
    __syncthreads();

    const float wz = (z == 0 || z == 127) ? 2.0f : 3.0f;
    float ovh = 0.0f, surf = 0.0f, occ = 0.0f;
    for (int e = tid; e < HW; e += 256) {
        const int y = e >> 7, x = e & 127;
        const float cur = vox[base + e];
        occ += cur;
        const float wy = (y == 0 || y == 127) ? 2.0f : 3.0f;
        const float wx = (x == 0 || x == 127) ? 2.0f : 3.0f;
        surf += cur * (wz * wy * wx);
        if (z > 0) {
            float box = 0.0f;
#pragma unroll
            for (int dy = -1; dy <= 1; ++dy) {
                const int yy = y + dy;
                if (yy < 0 || yy > 127) continue;
#pragma unroll
                for (int dx = -1; dx <= 1; ++dx) {
                    const int xx = x + dx;
                    if (xx < 0 || xx > 127) continue;
                    box += sm[yy * 128 + xx];
                }
            }
            ovh += cur * (1.0f - box * (1.0f / 9.0f));
        }
    }
    __syncthreads();                      // done reading slice; reuse LDS
    sm[tid] = ovh; sm[256 + tid] = surf; sm[512 + tid] = occ;
    for (int off = 128; off > 0; off >>= 1) {
        __syncthreads();
        if (tid < off) {
            sm[tid]       += sm[tid + off];
            sm[256 + tid] += sm[256 + tid + off];
            sm[512 + tid] += sm[512 + tid + off];
        }
    }
    if (tid == 0) { p_ovh[s] = sm[0]; p_surf[s] = sm[256]; p_occ[s] = sm[512]; }
}

// ===========================================================================
// Kernel 2: mask = seed (1 at (b,5,5,5), else 0)
// ===========================================================================
__global__ __launch_bounds__(256)
void seed_init_kernel(float* __restrict__ mask) {
    const long long idx = (long long)blockIdx.x * 256 + threadIdx.x;
    const long long r = idx & (DHW - 1);                 // DHW = 2^21
    mask[idx] = (r == (5 * HW + 5 * 128 + 5)) ? 1.0f : 0.0f;
}

// ===========================================================================
// Kernel 3: one resin-trap sweep. 1024 threads per (b,z) slice; slice of the
// source mask is TDM-staged to LDS. Each wave owns 16x16 tiles; the in-plane
// 4-neighbor sum is D = T*M + M*T (T = tridiagonal ones band matrix) via
// 8 x V_WMMA_F32_16X16X4_F32; tile-edge halo and z+-1 terms fixed up in VALU.
// D layout: VGPR r, lanes 0-15 -> (M=r, N=lane), lanes 16-31 -> (M=r+8).
// ===========================================================================
__global__ __launch_bounds__(1024)
void resin_iter_kernel(const float* __restrict__ src,
                       const float* __restrict__ vox,
                       float* __restrict__ dst) {
    __shared__ float sm[HW];
    const int s   = blockIdx.x;
    const int z   = s & 127;
    const int tid = threadIdx.x;
    const long long base = (long long)s * HW;

    load_slice_to_lds(sm, src + base, tid);
    __syncthreads();

    const int lane = tid & 31;
    const int wave = tid >> 5;
    const int half = lane >> 4;          // which 16-lane half
    const int l    = lane & 15;

    for (int t = wave; t < 64; t += 32) {            // 64 tiles, 2 per wave
        const int Y0 = (t >> 3) << 4;
        const int X0 = (t & 7) << 4;
        v8f acc = {0.f, 0.f, 0.f, 0.f, 0.f, 0.f, 0.f, 0.f};

        // vertical neighbors: acc += T x M   (A = T cols 4k..4k+3, B = M rows)
        // T[i][j] = (|i-j| == 1)
#pragma unroll
        for (int k = 0; k < 4; ++k) {
            const int c0 = 4 * k + 2 * half;
            v2f a, b;
            a.x = ((l == c0 - 1) || (l == c0 + 1)) ? 1.0f : 0.0f;      // T[l][c0]
            a.y = ((l == c0)     || (l == c0 + 2)) ? 1.0f : 0.0f;      // T[l][c0+1]
            b.x = sm[(Y0 + c0)     * 128 + X0 + l];                    // M[c0][l]
            b.y = sm[(Y0 + c0 + 1) * 128 + X0 + l];                    // M[c0+1][l]
            acc = __builtin_amdgcn_wmma_f32_16x16x4_f32(false, a, false, b,
                                                        (short)0, acc, false, false);
        }
        // horizontal neighbors: acc += M x T (A = M cols, B = T rows)
#pragma unroll
        for (int k = 0; k < 4; ++k) {
            const int c0 = 4 * k + 2 * half;
            v2f a, b;
            a.x = sm[(Y0 + l) * 128 + X0 + c0];                        // M[l][c0]
            a.y = sm[(Y0 + l) * 128 + X0 + c0 + 1];                    // M[l][c0+1]
            b.x = ((l == c0 - 1) || (l == c0 + 1)) ? 1.0f : 0.0f;      // T[c0][l]
            b.y = ((l == c0)     || (l == c0 + 2)) ? 1.0f : 0.0f;      // T[c0+1][l]
            acc = __builtin_amdgcn_wmma_f32_16x16x4_f32(false, a, false, b,
                                                        (short)0, acc, false, false);
        }

#pragma unroll
        for (int r = 0; r < 8; ++r) {
            const int m = r + 8 * half;
            const int y = Y0 + m, x = X0 + l;
            float c = acc[r];
            // halo from adjacent tiles (zero padding at slice boundary)
            if (m == 0  && y > 0)   c += sm[(y - 1) * 128 + x];
            if (m == 15 && y < 127) c += sm[(y + 1) * 128 + x];
            if (l == 0  && x > 0)   c += sm[y * 128 + x - 1];
            if (l == 15 && x < 127) c += sm[y * 128 + x + 1];
            // z +- 1 neighbors (L2-resident)
            const long long p = base + y * 128 + x;
            if (z > 0)   c += src[p - HW];
            if (z < 127) c += src[p + HW];
            // sigmoid(20*(c-0.5)) = 1/(1+exp(10-20c))
            const float act = 1.0f / (1.0f + __expf(10.0f - 20.0f * c));
            const float old = sm[y * 128 + x];
            dst[p] = fmaxf(old, act) * (1.0f - vox[p]);
        }
    }
}

// ===========================================================================
// Kernel 4: per-slice sum of the final mask (fixed-order -> deterministic)
// ===========================================================================
__global__ __launch_bounds__(256)
void slice_sum_kernel(const float* __restrict__ m, float* __restrict__ part) {
    __shared__ float red[256];
    const int s = blockIdx.x, tid = threadIdx.x;
    const long long base = (long long)s * HW;
    float acc = 0.0f;
    for (int e = tid; e < HW; e += 256) acc += m[base + e];
    red[tid] = acc;
    for (int off = 128; off > 0; off >>= 1) {
        __syncthreads();
        if (tid < off) red[tid] += red[tid + off];
    }
    if (tid == 0) part[s] = red[0];
}

// ===========================================================================
// Kernel 5: combine 512 partials per quantity in fixed order, emit outputs
// ===========================================================================
__global__ void finalize_kernel(const float* __restrict__ p_ovh,
                                const float* __restrict__ p_surf,
                                const float* __restrict__ p_occ,
                                const float* __restrict__ p_enc,
                                float* __restrict__ out) {
    if (threadIdx.x == 0 && blockIdx.x == 0) {
        float so = 0.f, ss = 0.f, sc = 0.f, se = 0.f;
        for (int i = 0; i < NSLICE; ++i) {
            so += p_ovh[i]; ss += p_surf[i]; sc += p_occ[i]; se += p_enc[i];
        }
        const float invN     = 1.0f / (float)NTOT;
        const float overhang = so * invN;
        const float surface  = ss * (1.0f / 27.0f) * invN;
        const float occ      = sc * invN;
        const float occ_pen  = 10.0f * (occ - 0.5f) * (occ - 0.5f);
        const float resin    = 100.0f * se * invN;
        out[0] = overhang;
        out[1] = surface - resin - occ_pen;
    }
}

// ===========================================================================
extern "C" void kernel_launch(void* const* d_in, const int* in_sizes, int n_in,
                              void* d_out, int out_size, void* d_ws, size_t ws_size,
                              hipStream_t stream) {
    const float* vox = (const float*)d_in[0];
    float* out = (float*)d_out;

    char* ws = (char*)d_ws;
    float* maskA = (float*)(ws);
    float* maskB = (float*)(ws + (size_t)NTOT * 4);
    float* parts = (float*)(ws + (size_t)NTOT * 8);
    float* p_ovh  = parts;
    float* p_surf = parts + NSLICE;
    float* p_occ  = parts + 2 * NSLICE;
    float* p_enc  = parts + 3 * NSLICE;

    metrics_kernel<<<NSLICE, 256, 0, stream>>>(vox, p_ovh, p_surf, p_occ);
    seed_init_kernel<<<NTOT / 256, 256, 0, stream>>>(maskA);
    for (int i = 0; i < 10; ++i) {
        const float* src = (i & 1) ? maskB : maskA;
        float*       dst = (i & 1) ? maskA : maskB;
        resin_iter_kernel<<<NSLICE, 1024, 0, stream>>>(src, vox, dst);
    }
    // after i=9 (odd), final mask is in maskA
    slice_sum_kernel<<<NSLICE, 256, 0, stream>>>(maskA, p_enc);
    finalize_kernel<<<1, 32, 0, stream>>>(p_ovh, p_surf, p_occ, p_enc, out);
}